// RETA_43353399886386
// MI455X (gfx1250) — compile-verified
//
#include <hip/hip_runtime.h>

typedef float v2f __attribute__((ext_vector_type(2)));
typedef float v8f __attribute__((ext_vector_type(8)));

#define EDIM 128
#define WDIM 126
#define ROWP 132          // padded LDS row (max read idx = 112+15+2 = 129)
#define ZROW 12           // extra all-zero row for K-padding taps
#define NROWS 13
#define NCH  256          // 64 bn1 channels + 192 bn2 (a,f) channels
#define BATCH 4096

// ---- shared helpers -------------------------------------------------------

// Per-lane constant LDS word offset of tap (ks, v) for image `im`.
// A-fragment layout for V_WMMA_F32_16X16X4_F32:
// lanes 0-15: M=lane, K = {0,1} in v0/v1 ; lanes 16-31: M=lane-16, K = {2,3}
// Padding taps (t >= 9) point at the zeroed ZROW.
__device__ __forceinline__ int a_tap_off(int im, int ks, int v, int khalf,
                                         int mrow) {
    int t = ks * 4 + khalf * 2 + v;              // tap index, K padded to 12
    int i = t / 3, j = t - 3 * i;                // 3x3 kernel position (t<9)
    return (t < 9) ? ((im * 3 + i) * ROWP + j + mrow) : (ZROW * ROWP + mrow);
}

// B fragment (4x16, K x N): N = lane%16, K split like A across lane halves
__device__ __forceinline__ v2f build_b_frag(const float* wmat, int f, int ks,
                                            int khalf) {
    v2f b;
#pragma unroll
    for (int v = 0; v < 2; ++v) {
        int t = ks * 4 + khalf * 2 + v;
        b[v] = (t < 9) ? wmat[f * 9 + t] : 0.f;
    }
    return b;
}

// Gather the 12 image rows (4 images x 3 rows of 3x3 conv input) into LDS,
// plus one all-zero row used by the K-padding taps.
__device__ __forceinline__ void gather_rows(float* lds, const int* xb,
                                            const float* Rel, const float* Ent,
                                            const float* Typ, int b, int tid) {
    int c[10];
#pragma unroll
    for (int k = 0; k < 10; ++k) c[k] = xb[b * 10 + k];
    const float* r0 = Rel + (size_t)c[0] * EDIM;
    const float* srcs[12] = {
        Ent + (size_t)c[1] * EDIM, r0, Ent + (size_t)c[3] * EDIM,   // img0: e0,r0,e1
        Typ + (size_t)c[4] * EDIM, r0, Typ + (size_t)c[5] * EDIM,   // img1: h0,r0,t0
        Typ + (size_t)c[6] * EDIM, r0, Typ + (size_t)c[7] * EDIM,   // img2
        Typ + (size_t)c[8] * EDIM, r0, Typ + (size_t)c[9] * EDIM }; // img3
    if (tid < ROWP) {
#pragma unroll
        for (int row = 0; row < 12; ++row) {     // row is a constant after unroll
            const float* src = srcs[row];
            lds[row * ROWP + tid] = (tid < EDIM) ? src[tid] : 0.f;
        }
        lds[ZROW * ROWP + tid] = 0.f;
    }
}

// ---- pass A: conv (WMMA) + deterministic per-b channel stats --------------

__global__ __launch_bounds__(256)
void reta_conv_stats(const int* __restrict__ xb, const float* __restrict__ Rel,
                     const float* __restrict__ Ent, const float* __restrict__ Typ,
                     const float* __restrict__ w1, const float* __restrict__ cb1,
                     const float* __restrict__ w2, const float* __restrict__ cb2,
                     float* __restrict__ part) {
    __shared__ float lds[NROWS * ROWP];
    int b = blockIdx.x, tid = threadIdx.x;
    gather_rows(lds, xb, Rel, Ent, Typ, b, tid);
    __syncthreads();

    int lane = tid & 31, wave = tid >> 5;
    int im = wave >> 1, nhalf = wave & 1;        // wave owns (image, channel-half)
    int n = lane & 15, khalf = lane >> 4, mrow = n;

    const float* wmat = (im == 0) ? w1 : w2;
    const float* cbv  = (im == 0) ? cb1 : cb2;

    // loop-invariant per-lane A-tap addresses (word offsets into LDS)
    int aoff[3][2];
#pragma unroll
    for (int ks = 0; ks < 3; ++ks)
#pragma unroll
        for (int v = 0; v < 2; ++v)
            aoff[ks][v] = a_tap_off(im, ks, v, khalf, mrow);

    v2f bf[2][3];
    float bias[2];
#pragma unroll
    for (int nt = 0; nt < 2; ++nt) {
        int f = (nhalf * 2 + nt) * 16 + n;
        bias[nt] = cbv[f];
#pragma unroll
        for (int ks = 0; ks < 3; ++ks) bf[nt][ks] = build_b_frag(wmat, f, ks, khalf);
    }

    float s[2] = {0.f, 0.f}, q[2] = {0.f, 0.f};
#pragma unroll
    for (int xt = 0; xt < 8; ++xt) {
        int x0 = xt * 16;
        v2f af[3];
#pragma unroll
        for (int ks = 0; ks < 3; ++ks) {
            af[ks][0] = lds[aoff[ks][0] + x0];
            af[ks][1] = lds[aoff[ks][1] + x0];
        }
#pragma unroll
        for (int nt = 0; nt < 2; ++nt) {
            v8f c = {};
#pragma unroll
            for (int ks = 0; ks < 3; ++ks)
                c = __builtin_amdgcn_wmma_f32_16x16x4_f32(false, af[ks], false,
                                                          bf[nt][ks], (short)0, c,
                                                          false, false);
#pragma unroll
            for (int v = 0; v < 8; ++v) {
                int x = x0 + v + 8 * khalf;      // D layout: m = v + 8*(lane>=16)
                float dv = c[v] + bias[nt];
                bool ok = x < WDIM;
                s[nt] += ok ? dv : 0.f;
                q[nt] += ok ? dv * dv : 0.f;
            }
        }
    }
#pragma unroll
    for (int nt = 0; nt < 2; ++nt) {
        s[nt] += __shfl_xor(s[nt], 16, 32);
        q[nt] += __shfl_xor(q[nt], 16, 32);
    }
    if (lane < 16) {
#pragma unroll
        for (int nt = 0; nt < 2; ++nt) {
            int f = (nhalf * 2 + nt) * 16 + n;
            int ch = (im == 0) ? f : (64 + (im - 1) * 64 + f);
            size_t base = (size_t)b * 512 + (size_t)ch * 2;
            part[base]     = s[nt];
            part[base + 1] = q[nt];
        }
    }
}

// ---- reduce stats over b -> per-channel scale/shift -----------------------

__global__ __launch_bounds__(128)
void reta_stats_finalize(const float* __restrict__ part,
                         const float* __restrict__ g1, const float* __restrict__ be1,
                         const float* __restrict__ g2, const float* __restrict__ be2,
                         float* __restrict__ ss) {
    __shared__ float rs[128], rq[128];
    int ch = blockIdx.x, tid = threadIdx.x;
    float S = 0.f, Q = 0.f;
    for (int b = tid; b < BATCH; b += 128) {
        size_t base = (size_t)b * 512 + (size_t)ch * 2;
        S += part[base];
        Q += part[base + 1];
    }
    rs[tid] = S; rq[tid] = Q;
    __syncthreads();
    for (int off = 64; off > 0; off >>= 1) {
        if (tid < off) { rs[tid] += rs[tid + off]; rq[tid] += rq[tid + off]; }
        __syncthreads();
    }
    if (tid == 0) {
        const float cnt = (float)BATCH * (float)WDIM;   // 516096 for bn1 and bn2
        float mean = rs[0] / cnt;
        float var = fmaxf(rq[0] / cnt - mean * mean, 0.f);
        float g, be;
        if (ch < 64) { g = g1[ch];  be = be1[ch]; }
        else         { int f = (ch - 64) & 63; g = g2[f]; be = be2[f]; }
        float sc = g * rsqrtf(var + 1e-5f);
        ss[ch * 2]     = sc;
        ss[ch * 2 + 1] = be - mean * sc;
    }
}

// ---- pass B: conv again + bn/relu/min + FC dot ----------------------------

__global__ __launch_bounds__(256)
void reta_apply(const int* __restrict__ xb, const float* __restrict__ Rel,
                const float* __restrict__ Ent, const float* __restrict__ Typ,
                const float* __restrict__ w1, const float* __restrict__ cb1,
                const float* __restrict__ w2, const float* __restrict__ cb2,
                const float* __restrict__ ss, const float* __restrict__ fw,
                const float* __restrict__ fb, float* __restrict__ out) {
    __shared__ float lds[NROWS * ROWP];
    __shared__ float red[256];
    int b = blockIdx.x, tid = threadIdx.x;
    gather_rows(lds, xb, Rel, Ent, Typ, b, tid);
    __syncthreads();

    int lane = tid & 31, wave = tid >> 5;
    int ntile = wave & 3, xh = wave >> 2;        // wave owns 1 channel tile, 4 x-tiles
    int n = lane & 15, khalf = lane >> 4, mrow = n;
    int f = ntile * 16 + n;

    // loop-invariant per-lane A-tap addresses for all 4 images
    int aoff[4][3][2];
#pragma unroll
    for (int im = 0; im < 4; ++im)
#pragma unroll
        for (int ks = 0; ks < 3; ++ks)
#pragma unroll
            for (int v = 0; v < 2; ++v)
                aoff[im][ks][v] = a_tap_off(im, ks, v, khalf, mrow);

    v2f bf1[3], bf2[3];
#pragma unroll
    for (int ks = 0; ks < 3; ++ks) {
        bf1[ks] = build_b_frag(w1, f, ks, khalf);
        bf2[ks] = build_b_frag(w2, f, ks, khalf);
    }
    float bias1 = cb1[f], bias2 = cb2[f];
    float sc1 = ss[f * 2], sh1 = ss[f * 2 + 1];
    float sc2[3], sh2[3];
#pragma unroll
    for (int a = 0; a < 3; ++a) {
        int ch = 64 + a * 64 + f;
        sc2[a] = ss[ch * 2];
        sh2[a] = ss[ch * 2 + 1];
    }

    float acc = 0.f;
#pragma unroll
    for (int xr = 0; xr < 4; ++xr) {
        int xt = xh * 4 + xr;
        int x0 = xt * 16;
        v8f c[4];
#pragma unroll
        for (int im = 0; im < 4; ++im) {
            v2f af[3];
#pragma unroll
            for (int ks = 0; ks < 3; ++ks) {
                af[ks][0] = lds[aoff[im][ks][0] + x0];
                af[ks][1] = lds[aoff[im][ks][1] + x0];
            }
            v8f cc = {};
#pragma unroll
            for (int ks = 0; ks < 3; ++ks) {
                v2f bb = (im == 0) ? bf1[ks] : bf2[ks];
                cc = __builtin_amdgcn_wmma_f32_16x16x4_f32(false, af[ks], false, bb,
                                                           (short)0, cc, false, false);
            }
            c[im] = cc;
        }
#pragma unroll
        for (int v = 0; v < 8; ++v) {
            int x = x0 + v + 8 * khalf;
            if (x < WDIM) {
                float z1 = fmaxf((c[0][v] + bias1) * sc1 + sh1, 0.f);
                acc += z1 * fw[f * WDIM + x];
                float mn = 3.4e38f;
#pragma unroll
                for (int a = 0; a < 3; ++a) {
                    float z = fmaxf((c[a + 1][v] + bias2) * sc2[a] + sh2[a], 0.f);
                    mn = fminf(mn, z);
                }
                acc += mn * fw[64 * WDIM + f * WDIM + x];
            }
        }
    }
    red[tid] = acc;
    __syncthreads();
    for (int off = 128; off > 0; off >>= 1) {
        if (tid < off) red[tid] += red[tid + off];
        __syncthreads();
    }
    if (tid == 0) out[b] = red[0] + fb[0];
}

// ---- launch ---------------------------------------------------------------

extern "C" void kernel_launch(void* const* d_in, const int* in_sizes, int n_in,
                              void* d_out, int out_size, void* d_ws, size_t ws_size,
                              hipStream_t stream) {
    const int*   xb  = (const int*)  d_in[0];
    const float* Rel = (const float*)d_in[3];
    const float* Ent = (const float*)d_in[4];
    const float* Typ = (const float*)d_in[5];
    const float* w1  = (const float*)d_in[6];
    const float* cb1 = (const float*)d_in[7];
    const float* g1  = (const float*)d_in[8];
    const float* be1 = (const float*)d_in[9];
    const float* w2  = (const float*)d_in[10];
    const float* cb2 = (const float*)d_in[11];
    const float* g2  = (const float*)d_in[12];
    const float* be2 = (const float*)d_in[13];
    const float* fw  = (const float*)d_in[14];
    const float* fb  = (const float*)d_in[15];
    float* out  = (float*)d_out;
    float* part = (float*)d_ws;                     // BATCH * 512 floats (8 MB)
    float* ss   = part + (size_t)BATCH * 512;       // NCH * 2 floats

    reta_conv_stats<<<BATCH, 256, 0, stream>>>(xb, Rel, Ent, Typ, w1, cb1, w2, cb2, part);
    reta_stats_finalize<<<NCH, 128, 0, stream>>>(part, g1, be1, g2, be2, ss);
    reta_apply<<<BATCH, 256, 0, stream>>>(xb, Rel, Ent, Typ, w1, cb1, w2, cb2, ss, fw, fb, out);
}